// SentenceDecoder_9448928051413
// MI455X (gfx1250) — compile-verified
//
#include <hip/hip_runtime.h>
#include <math.h>

// Problem constants
#define B_   32
#define R_   256
#define E_   512
#define H_   512
#define V_   10000
#define T_   32
#define VP_  16
#define L_   16
#define BOS_ 1
#define INSZ_ 1568   // E + VP + L + H + E
#define KC_   1056   // constant part of x: E + VP + L + H

typedef __attribute__((ext_vector_type(2))) float v2f;
typedef __attribute__((ext_vector_type(8))) float v8f;

__device__ __forceinline__ float sigmoidf_(float x) { return 1.0f / (1.0f + expf(-x)); }

// ---------------------------------------------------------------------------
// Register-tiled WMMA f32 GEMM:
//   C[M,N] = act( A[M,K](lda) @ W[N,K](ldw)^T + biases (+ Cin) )
// (MT+NT) b64 loads feed MT*NT WMMAs per K-step.
// ACT: 0=none 1=relu 2=tanh ; NBIAS: 0,1,2 ; CIN: add Cin matrix in epilogue
// ---------------------------------------------------------------------------
template <int MT, int NT, int ACT, int NBIAS, int CIN>
__global__ void wmma_gemm_xwt(const float* __restrict__ A, int lda,
                              const float* __restrict__ W, int ldw,
                              const float* __restrict__ bias0,
                              const float* __restrict__ bias1,
                              const float* __restrict__ Cin,
                              float* __restrict__ C, int ldc,
                              int M, int N, int K)
{
    int wave = (blockIdx.x * blockDim.x + threadIdx.x) >> 5;
    int lane = threadIdx.x & 31;
    int nwt = N / (16 * NT);
    int mwt = M / (16 * MT);
    if (wave >= nwt * mwt) return;        // wave-uniform: EXEC stays all-ones
    int mw = wave / nwt, nw = wave % nwt;
    int half = lane >> 4;                 // 0: K pair {0,1}, 1: K pair {2,3}
    int l15  = lane & 15;
    int m0 = mw * 16 * MT;
    int n0 = nw * 16 * NT;

    const float* Arow[MT];
    const float* Wrow[NT];
#pragma unroll
    for (int i = 0; i < MT; ++i)
        Arow[i] = A + (size_t)(m0 + 16 * i + l15) * lda + 2 * half;
#pragma unroll
    for (int j = 0; j < NT; ++j)
        Wrow[j] = W + (size_t)(n0 + 16 * j + l15) * ldw + 2 * half;

    v8f acc[MT][NT] = {};

    for (int k = 0; k < K; k += 4) {
        v2f a[MT], b[NT];
#pragma unroll
        for (int i = 0; i < MT; ++i) a[i] = *(const v2f*)(Arow[i] + k);
#pragma unroll
        for (int j = 0; j < NT; ++j) b[j] = *(const v2f*)(Wrow[j] + k);
#pragma unroll
        for (int i = 0; i < MT; ++i)
#pragma unroll
            for (int j = 0; j < NT; ++j)
                acc[i][j] = __builtin_amdgcn_wmma_f32_16x16x4_f32(
                    false, a[i], false, b[j], (short)0, acc[i][j], false, false);
    }

#pragma unroll
    for (int j = 0; j < NT; ++j) {
        int n = n0 + 16 * j + l15;
        float bsum = 0.0f;
        if (NBIAS >= 1) bsum += bias0[n];
        if (NBIAS >= 2) bsum += bias1[n];
#pragma unroll
        for (int i = 0; i < MT; ++i) {
#pragma unroll
            for (int r = 0; r < 8; ++r) {
                int m = m0 + 16 * i + r + 8 * half;
                float v = acc[i][j][r] + bsum;
                if (CIN) v += Cin[(size_t)m * ldc + n];
                if (ACT == 1) v = fmaxf(v, 0.0f);
                if (ACT == 2) v = tanhf(v);
                C[(size_t)m * ldc + n] = v;
            }
        }
    }
}

// ---------------------------------------------------------------------------
// Fused LSTM-gates GEMM with embedded token gather:
//   g = emb[tok] @ Wih[:,1056:]^T + h @ Whh^T + gconst
// M = 32 (MT=2 covers all of M -> every weight element loaded exactly once),
// N = 2048 (NT=4), K = 512 per pass. 32 wave-jobs.
// ---------------------------------------------------------------------------
__global__ void gates_kernel(const int* __restrict__ text, const float* __restrict__ emb,
                             int t,
                             const float* __restrict__ Wih_te,   // Wih + KC_, ldw = INSZ_
                             const float* __restrict__ h,
                             const float* __restrict__ Whh,
                             const float* __restrict__ gconst,
                             float* __restrict__ g)
{
    const int MT = 2, NT = 4;
    int wave = (blockIdx.x * blockDim.x + threadIdx.x) >> 5;
    int lane = threadIdx.x & 31;
    const int nwt = (4 * H_) / (16 * NT);   // 32
    if (wave >= nwt) return;
    int half = lane >> 4, l15 = lane & 15;
    int n0 = wave * 16 * NT;

    const float* A1[MT];
    const float* A2[MT];
    const float* W1[NT];
    const float* W2[NT];
#pragma unroll
    for (int i = 0; i < MT; ++i) {
        int m = 16 * i + l15;
        int tok = (t == 0) ? BOS_ : text[m * T_ + (t - 1)];
        A1[i] = emb + (size_t)tok * E_ + 2 * half;    // embedding gather fused
        A2[i] = h + (size_t)m * H_ + 2 * half;
    }
#pragma unroll
    for (int j = 0; j < NT; ++j) {
        int n = n0 + 16 * j + l15;
        W1[j] = Wih_te + (size_t)n * INSZ_ + 2 * half;
        W2[j] = Whh + (size_t)n * H_ + 2 * half;
    }

    v8f acc[MT][NT] = {};
    for (int k = 0; k < E_; k += 4) {
        v2f a[MT], b[NT];
#pragma unroll
        for (int i = 0; i < MT; ++i) a[i] = *(const v2f*)(A1[i] + k);
#pragma unroll
        for (int j = 0; j < NT; ++j) b[j] = *(const v2f*)(W1[j] + k);
#pragma unroll
        for (int i = 0; i < MT; ++i)
#pragma unroll
            for (int j = 0; j < NT; ++j)
                acc[i][j] = __builtin_amdgcn_wmma_f32_16x16x4_f32(
                    false, a[i], false, b[j], (short)0, acc[i][j], false, false);
    }
    for (int k = 0; k < H_; k += 4) {
        v2f a[MT], b[NT];
#pragma unroll
        for (int i = 0; i < MT; ++i) a[i] = *(const v2f*)(A2[i] + k);
#pragma unroll
        for (int j = 0; j < NT; ++j) b[j] = *(const v2f*)(W2[j] + k);
#pragma unroll
        for (int i = 0; i < MT; ++i)
#pragma unroll
            for (int j = 0; j < NT; ++j)
                acc[i][j] = __builtin_amdgcn_wmma_f32_16x16x4_f32(
                    false, a[i], false, b[j], (short)0, acc[i][j], false, false);
    }

#pragma unroll
    for (int j = 0; j < NT; ++j) {
        int n = n0 + 16 * j + l15;
#pragma unroll
        for (int i = 0; i < MT; ++i) {
#pragma unroll
            for (int r = 0; r < 8; ++r) {
                int m = 16 * i + r + 8 * half;
                g[(size_t)m * (4 * H_) + n] = acc[i][j][r] + gconst[(size_t)m * (4 * H_) + n];
            }
        }
    }
}

// ---------------------------------------------------------------------------
// Two independent M=32, N=512, K=512 GEMMs in one launch (waves split evenly;
// the split is wave-uniform so EXEC stays all-ones for WMMA).
// GEMMi: Ci = act(Ai @ Wi^T + bi), ldw = ldc = 512.
// ---------------------------------------------------------------------------
template <int ACT>
__global__ void wmma_gemm_dual(const float* __restrict__ A0, int lda0,
                               const float* __restrict__ W0,
                               const float* __restrict__ b0,
                               float* __restrict__ C0,
                               const float* __restrict__ A1, int lda1,
                               const float* __restrict__ W1,
                               const float* __restrict__ b1,
                               float* __restrict__ C1)
{
    const int MT = 2, NT = 4;
    const int nwt = H_ / (16 * NT);   // 8 wave-jobs per GEMM
    int wave = (blockIdx.x * blockDim.x + threadIdx.x) >> 5;
    int lane = threadIdx.x & 31;
    if (wave >= 2 * nwt) return;
    bool second = wave >= nwt;
    const float* A = second ? A1 : A0;
    int lda = second ? lda1 : lda0;
    const float* W = second ? W1 : W0;
    const float* bias = second ? b1 : b0;
    float* C = second ? C1 : C0;
    int nw = second ? (wave - nwt) : wave;

    int half = lane >> 4, l15 = lane & 15;
    int n0 = nw * 16 * NT;

    const float* Arow[MT];
    const float* Wrow[NT];
#pragma unroll
    for (int i = 0; i < MT; ++i)
        Arow[i] = A + (size_t)(16 * i + l15) * lda + 2 * half;
#pragma unroll
    for (int j = 0; j < NT; ++j)
        Wrow[j] = W + (size_t)(n0 + 16 * j + l15) * H_ + 2 * half;

    v8f acc[MT][NT] = {};
    for (int k = 0; k < H_; k += 4) {
        v2f a[MT], b[NT];
#pragma unroll
        for (int i = 0; i < MT; ++i) a[i] = *(const v2f*)(Arow[i] + k);
#pragma unroll
        for (int j = 0; j < NT; ++j) b[j] = *(const v2f*)(Wrow[j] + k);
#pragma unroll
        for (int i = 0; i < MT; ++i)
#pragma unroll
            for (int j = 0; j < NT; ++j)
                acc[i][j] = __builtin_amdgcn_wmma_f32_16x16x4_f32(
                    false, a[i], false, b[j], (short)0, acc[i][j], false, false);
    }

#pragma unroll
    for (int j = 0; j < NT; ++j) {
        int n = n0 + 16 * j + l15;
        float bsum = bias[n];
#pragma unroll
        for (int i = 0; i < MT; ++i) {
#pragma unroll
            for (int r = 0; r < 8; ++r) {
                int m = 16 * i + r + 8 * half;
                float v = acc[i][j][r] + bsum;
                if (ACT == 2) v = tanhf(v);
                C[(size_t)m * H_ + n] = v;
            }
        }
    }
}

// ---------------------------------------------------------------------------
// Small helper kernels
// ---------------------------------------------------------------------------
__global__ void mean_kernel(const float* __restrict__ image, float* __restrict__ imean)
{
    int idx = blockIdx.x * blockDim.x + threadIdx.x;
    if (idx >= B_ * E_) return;
    int b = idx / E_, e = idx % E_;
    const float* p = image + (size_t)b * R_ * E_ + e;
    float s = 0.0f;
    for (int r = 0; r < R_; ++r) s += p[(size_t)r * E_];
    imean[idx] = s * (1.0f / R_);
}

__global__ void xconst_kernel(const float* __restrict__ imean,
                              const float* __restrict__ vp,
                              const float* __restrict__ lab,
                              const float* __restrict__ topic,
                              float* __restrict__ xc)
{
    int idx = blockIdx.x * blockDim.x + threadIdx.x;
    if (idx >= B_ * KC_) return;
    int b = idx / KC_, c = idx % KC_;
    float v;
    if (c < E_)                 v = imean[b * E_ + c];
    else if (c < E_ + VP_)      v = vp[b * VP_ + (c - E_)];
    else if (c < E_ + VP_ + L_) v = lab[b * L_ + (c - E_ - VP_)];
    else                        v = topic[b * H_ + (c - E_ - VP_ - L_)];
    xc[idx] = v;
}

__global__ void lstm_kernel(const float* __restrict__ g,
                            float* __restrict__ h, float* __restrict__ m)
{
    int idx = blockIdx.x * blockDim.x + threadIdx.x;
    if (idx >= B_ * H_) return;
    int b = idx / H_, j = idx % H_;
    const float* gb = g + (size_t)b * 4 * H_;
    float ig = sigmoidf_(gb[j]);
    float fg = sigmoidf_(gb[H_ + j]);
    float gg = tanhf(gb[2 * H_ + j]);
    float og = sigmoidf_(gb[3 * H_ + j]);
    float mn = fg * m[idx] + ig * gg;
    float hn = og * tanhf(mn);
    m[idx] = mn;
    h[idx] = hn;
}

// One wave per (b, r): z[b,r] = sum_e tanh(base[e] + _hh[b,e]) * Wz[e] + bz
__global__ void z_kernel(const float* __restrict__ vbuf, const float* __restrict__ ssp,
                         const float* __restrict__ hhp, const float* __restrict__ Wz,
                         const float* __restrict__ bz, float* __restrict__ z)
{
    int gid = blockIdx.x * blockDim.x + threadIdx.x;
    int wave = gid >> 5, lane = gid & 31;
    if (wave >= B_ * (R_ + 1)) return;
    int b = wave / (R_ + 1), r = wave % (R_ + 1);
    const float* base = (r < R_) ? (vbuf + ((size_t)b * R_ + r) * H_)
                                 : (ssp + (size_t)b * H_);
    const float* hh = hhp + (size_t)b * H_;
    float s = 0.0f;
    for (int e = lane; e < H_; e += 32) s += tanhf(base[e] + hh[e]) * Wz[e];
    for (int off = 16; off > 0; off >>= 1) s += __shfl_xor(s, off, 32);
    if (lane == 0) z[b * (R_ + 1) + r] = s + bz[0];
}

// One block per b: softmax over 257, write masked att, compute cph = c + hh
__global__ void attn_kernel(const float* __restrict__ z, const float* __restrict__ image,
                            const float* __restrict__ hs, const int* __restrict__ length,
                            int t, float* __restrict__ out_att, float* __restrict__ cph)
{
    __shared__ float sa[R_ + 1];
    __shared__ float red[256];
    int b = blockIdx.x, tid = threadIdx.x;
    const float* zb = z + b * (R_ + 1);

    float v = zb[tid];
    if (tid == 0) v = fmaxf(v, zb[R_]);
    red[tid] = v;
    __syncthreads();
    for (int s = 128; s > 0; s >>= 1) {
        if (tid < s) red[tid] = fmaxf(red[tid], red[tid + s]);
        __syncthreads();
    }
    float mx = red[0];
    __syncthreads();

    float e0 = expf(zb[tid] - mx);
    sa[tid] = e0;
    float loc = e0;
    if (tid == 0) { float e1 = expf(zb[R_] - mx); sa[R_] = e1; loc += e1; }
    red[tid] = loc;
    __syncthreads();
    for (int s = 128; s > 0; s >>= 1) {
        if (tid < s) red[tid] += red[tid + s];
        __syncthreads();
    }
    float inv = 1.0f / red[0];
    __syncthreads();
    sa[tid] *= inv;
    if (tid == 0) sa[R_] *= inv;
    __syncthreads();

    float mf = (t < length[b]) ? 1.0f : 0.0f;
    float* oa = out_att + ((size_t)b * T_ + t) * (R_ + 1);
    oa[tid] = sa[tid] * mf;
    if (tid == 0) oa[R_] = sa[R_] * mf;

    const float* img = image + (size_t)b * R_ * E_;
    const float* hsb = hs + (size_t)b * 2 * E_;
    for (int e = tid; e < E_; e += 256) {
        float c = 0.0f;
        for (int r = 0; r < R_; ++r) c += sa[r] * img[(size_t)r * E_ + e];
        c += sa[R_] * hsb[E_ + e];            // s contribution
        cph[(size_t)b * E_ + e] = c + hsb[e]; // + hh
    }
}

// One block (512 threads) per b: log_softmax(logits/temp) + argmax, masked write
__global__ void logsm_kernel(const float* __restrict__ logits, const float* __restrict__ temp,
                             const int* __restrict__ length, int t,
                             float* __restrict__ out_logp, float* __restrict__ out_txt)
{
    __shared__ float sv[512];
    __shared__ int   si[512];
    int b = blockIdx.x, tid = threadIdx.x;
    const float* lg = logits + (size_t)b * V_;

    float best = -INFINITY; int bi = 0;
    for (int i = tid; i < V_; i += 512) {
        float x = lg[i];
        if (x > best) { best = x; bi = i; }
    }
    sv[tid] = best; si[tid] = bi;
    __syncthreads();
    for (int s = 256; s > 0; s >>= 1) {
        if (tid < s) {
            float a = sv[tid], c = sv[tid + s];
            if (c > a || (c == a && si[tid + s] < si[tid])) { sv[tid] = c; si[tid] = si[tid + s]; }
        }
        __syncthreads();
    }
    float mx = sv[0];
    int am = si[0];
    __syncthreads();

    float tp = temp[b];
    float sum = 0.0f;
    for (int i = tid; i < V_; i += 512) sum += expf((lg[i] - mx) / tp);
    sv[tid] = sum;
    __syncthreads();
    for (int s = 256; s > 0; s >>= 1) {
        if (tid < s) sv[tid] += sv[tid + s];
        __syncthreads();
    }
    float lse = logf(sv[0]);

    bool msk = t < length[b];
    float mf = msk ? 1.0f : 0.0f;
    float* op = out_logp + ((size_t)b * T_ + t) * V_;
    for (int i = tid; i < V_; i += 512) op[i] = ((lg[i] - mx) / tp - lse) * mf;
    if (tid == 0) out_txt[b * T_ + t] = msk ? (float)am : 0.0f;
}

// ---------------------------------------------------------------------------
extern "C" void kernel_launch(void* const* d_in, const int* in_sizes, int n_in,
                              void* d_out, int out_size, void* d_ws, size_t ws_size,
                              hipStream_t stream)
{
    (void)in_sizes; (void)n_in; (void)out_size; (void)ws_size;

    const float* image  = (const float*)d_in[0];
    const float* viewp  = (const float*)d_in[1];
    const int*   text   = (const int*)  d_in[2];
    const float* label  = (const float*)d_in[3];
    const float* topic  = (const float*)d_in[4];
    const float* temp   = (const float*)d_in[5];
    const int*   length = (const int*)  d_in[6];
    const float* emb    = (const float*)d_in[7];
    const float* Wv  = (const float*)d_in[8],  *bv   = (const float*)d_in[9];
    const float* Wh  = (const float*)d_in[10], *bh   = (const float*)d_in[11];
    const float* Wm  = (const float*)d_in[12], *bm   = (const float*)d_in[13];
    const float* Wih = (const float*)d_in[14], *bih  = (const float*)d_in[15];
    const float* Whh = (const float*)d_in[16], *bhh  = (const float*)d_in[17];
    const float* Wfc = (const float*)d_in[18], *bfc  = (const float*)d_in[19];
    const float* Whh2= (const float*)d_in[20], *bhh2 = (const float*)d_in[21];
    const float* Ws  = (const float*)d_in[22], *bs   = (const float*)d_in[23];
    const float* Wz  = (const float*)d_in[24], *bz   = (const float*)d_in[25];
    const float* Wp  = (const float*)d_in[26], *bp   = (const float*)d_in[27];

    // Workspace layout (floats)
    float* ws = (float*)d_ws;
    float* imean  = ws; ws += B_ * E_;
    float* vbuf   = ws; ws += (size_t)B_ * R_ * H_;
    float* hbuf   = ws; ws += B_ * H_;
    float* mbuf   = ws; ws += B_ * H_;
    float* xc     = ws; ws += B_ * KC_;
    float* gconst = ws; ws += B_ * 4 * H_;
    float* gbuf   = ws; ws += B_ * 4 * H_;
    float* hsbuf  = ws; ws += B_ * 2 * E_;
    float* hhp    = ws; ws += B_ * H_;
    float* ssp    = ws; ws += B_ * H_;
    float* zbuf   = ws; ws += B_ * (R_ + 1);
    float* cph    = ws; ws += B_ * E_;
    float* logits = ws; ws += (size_t)B_ * V_;

    float* out_att  = (float*)d_out;
    float* out_logp = out_att + (size_t)B_ * T_ * (R_ + 1);
    float* out_txt  = out_logp + (size_t)B_ * T_ * V_;

    auto blocks_for = [](int M, int N, int MT, int NT) {
        int waves = (M / (16 * MT)) * (N / (16 * NT));
        return (waves + 7) / 8;
    };

    // ---- Precompute (hoisted out of the time loop) ----
    mean_kernel<<<(B_ * E_ + 255) / 256, 256, 0, stream>>>(image, imean);

    // v = image @ Wv^T + bv   (M = 8192, N = 512, K = 512): 64x32 tile/wave
    wmma_gemm_xwt<4, 2, 0, 1, 0><<<blocks_for(B_ * R_, H_, 4, 2), 256, 0, stream>>>(
        image, E_, Wv, E_, bv, nullptr, nullptr, vbuf, H_, B_ * R_, H_, E_);

    // h0 = tanh(imean @ Wh^T + bh) ; m0 = tanh(imean @ Wm^T + bm)  (one launch)
    wmma_gemm_dual<2><<<2, 256, 0, stream>>>(imean, E_, Wh, bh, hbuf,
                                             imean, E_, Wm, bm, mbuf);

    // gconst = [imean, vp, label, topic] @ Wih[:, :1056]^T + bih + bhh
    xconst_kernel<<<(B_ * KC_ + 255) / 256, 256, 0, stream>>>(imean, viewp, label, topic, xc);
    wmma_gemm_xwt<2, 4, 0, 2, 0><<<blocks_for(B_, 4 * H_, 2, 4), 256, 0, stream>>>(
        xc, KC_, Wih, INSZ_, bih, bhh, nullptr, gconst, 4 * H_, B_, 4 * H_, KC_);

    // ---- Time loop ----
    for (int t = 0; t < T_; ++t) {
        // g = emb[tok] @ Wih[:,1056:]^T + h @ Whh^T + gconst   (fused gather + dual-K)
        gates_kernel<<<4, 256, 0, stream>>>(text, emb, t, Wih + KC_, hbuf, Whh, gconst, gbuf);

        lstm_kernel<<<(B_ * H_ + 255) / 256, 256, 0, stream>>>(gbuf, hbuf, mbuf);

        // hs = relu(h @ Wfc^T + bfc)   (N = 1024)
        wmma_gemm_xwt<2, 4, 1, 1, 0><<<blocks_for(B_, 2 * E_, 2, 4), 256, 0, stream>>>(
            hbuf, H_, Wfc, H_, bfc, nullptr, nullptr, hsbuf, 2 * E_, B_, 2 * E_, H_);

        // _hh = hh @ Whh2^T + bhh2 ; _s = s @ Ws^T + bs   (one launch, wave split)
        wmma_gemm_dual<0><<<2, 256, 0, stream>>>(hsbuf, 2 * E_, Whh2, bhh2, hhp,
                                                 hsbuf + E_, 2 * E_, Ws, bs, ssp);

        z_kernel<<<(B_ * (R_ + 1) * 32 + 255) / 256, 256, 0, stream>>>(
            vbuf, ssp, hhp, Wz, bz, zbuf);

        attn_kernel<<<B_, 256, 0, stream>>>(zbuf, image, hsbuf, length, t, out_att, cph);

        // logits = cph @ Wp^T + bp   (N = 10000 -> 625 n-tiles = 125 * NT=5)
        wmma_gemm_xwt<2, 5, 0, 1, 0><<<blocks_for(B_, V_, 2, 5), 256, 0, stream>>>(
            cph, E_, Wp, E_, bp, nullptr, nullptr, logits, V_, B_, V_, E_);

        logsm_kernel<<<B_, 512, 0, stream>>>(logits, temp, length, t, out_logp, out_txt);
    }
}